// MambaSurv_72146860638522
// MI455X (gfx1250) — compile-verified
//
#include <hip/hip_runtime.h>
#include <math.h>

// ---------------------------------------------------------------------------
// Model dims (fixed by the reference)
// ---------------------------------------------------------------------------
#define BB   16
#define TT   1024
#define DIN  256
#define DDT  8
#define DD   256
#define NN   16
#define LL   4
#define BT   (BB * TT)

typedef float v2f __attribute__((ext_vector_type(2)));
typedef float v8f __attribute__((ext_vector_type(8)));

// ---------------------------------------------------------------------------
// Generic fp32 WMMA GEMM:  out[m][n] = sum_k A[m][k] * W[n][k] (+bias[n]) (+res[m][n])
// One wave (32 lanes) computes one 16x16 tile using V_WMMA_F32_16X16X4_F32.
//   A fragment (16x4, 2 VGPR): lane r=lane&15, half=lane>>4 -> A[r][k0+2*half +{0,1}]
//   B fragment (4x16, 2 VGPR): B[k][n] = W[n][k]            -> W[col0+r][k0+2*half +{0,1}]
//   C/D (8 VGPR): d[v] -> out[row0 + v + 8*half][col0 + r]
// ---------------------------------------------------------------------------
__global__ void __launch_bounds__(32)
gemm_wmma_f32(const float* __restrict__ A, int lda,
              const float* __restrict__ W, int ldw,
              const float* __restrict__ bias,
              const float* __restrict__ residual,
              float* __restrict__ out, int ldo,
              int K)
{
    const int lane = threadIdx.x;          // block is a single wave32
    const int r    = lane & 15;
    const int half = lane >> 4;
    const int row0 = blockIdx.x * 16;
    const int col0 = blockIdx.y * 16;

    const float* Arow = A + (size_t)(row0 + r) * lda;
    const float* Wrow = W + (size_t)(col0 + r) * ldw;

    v8f acc = {};
    #pragma unroll 8
    for (int k0 = 0; k0 < K; k0 += 4) {
        const int ka = k0 + 2 * half;                 // even -> 8B aligned
        v2f a = *(const v2f*)(Arow + ka);
        v2f b = *(const v2f*)(Wrow + ka);
        acc = __builtin_amdgcn_wmma_f32_16x16x4_f32(
                  /*neg_a=*/false, a, /*neg_b=*/false, b,
                  /*c_mod=*/(short)0, acc, /*reuse_a=*/false, /*reuse_b=*/false);
    }

    const float bn = bias ? bias[col0 + r] : 0.0f;
    #pragma unroll
    for (int v = 0; v < 8; ++v) {
        const int m = row0 + v + 8 * half;
        const size_t idx = (size_t)m * ldo + (col0 + r);
        float val = acc[v] + bn;
        if (residual) val += residual[idx];
        out[idx] = val;
    }
}

// ---------------------------------------------------------------------------
// Row LayerNorm over D=256 (+ optional exact GELU). One block (256 thr) per row.
// ---------------------------------------------------------------------------
__global__ void __launch_bounds__(256)
ln_act_kernel(const float* __restrict__ in,
              const float* __restrict__ g, const float* __restrict__ b,
              float* __restrict__ out, int do_gelu)
{
    __shared__ float s1[256];
    __shared__ float s2[256];
    const int row = blockIdx.x;
    const int tid = threadIdx.x;
    const float x = in[(size_t)row * DD + tid];
    s1[tid] = x;
    s2[tid] = x * x;
    __syncthreads();
    #pragma unroll
    for (int off = 128; off > 0; off >>= 1) {
        if (tid < off) { s1[tid] += s1[tid + off]; s2[tid] += s2[tid + off]; }
        __syncthreads();
    }
    const float m   = s1[0] * (1.0f / DD);
    const float var = s2[0] * (1.0f / DD) - m * m;
    float xn = (x - m) * rsqrtf(var + 1e-5f) * g[tid] + b[tid];
    if (do_gelu) xn = 0.5f * xn * (1.0f + erff(xn * 0.70710678118654752f));
    out[(size_t)row * DD + tid] = xn;
}

// ---------------------------------------------------------------------------
// delta = softplus(delta_t @ Wdt^T + bdt) * mask    (DDT=8 inner dot, VALU)
// ---------------------------------------------------------------------------
__global__ void __launch_bounds__(256)
delta_kernel(const float* __restrict__ dt,   // (B,T,8)
             const float* __restrict__ Wdt,  // (D,8)  (this layer)
             const float* __restrict__ bdt,  // (D)
             const float* __restrict__ mask, // (B,T)
             float* __restrict__ delta)      // (B,T,D)
{
    const int idx = blockIdx.x * 256 + threadIdx.x;   // over BT*D
    const int d  = idx & (DD - 1);
    const int bt = idx >> 8;
    float v = bdt[d];
    #pragma unroll
    for (int j = 0; j < DDT; ++j)
        v += dt[(size_t)bt * DDT + j] * Wdt[d * DDT + j];
    const float sp = (v > 20.0f) ? v : log1pf(expf(v));
    delta[idx] = sp * mask[bt];
}

// ---------------------------------------------------------------------------
// Selective scan, fused SiLU(z) gate.  Grid (B, D/64), 64 threads: one d-channel
// per thread, state s[16] in VGPRs, B_t / C_t broadcast through LDS.
// ---------------------------------------------------------------------------
__global__ void __launch_bounds__(64)
scan_kernel(const float* __restrict__ xz,    // (B,T,2D)  [x | z]
            const float* __restrict__ delta, // (B,T,D)
            const float* __restrict__ Bm,    // (B,T,N)
            const float* __restrict__ Cm,    // (B,T,N)
            const float* __restrict__ A_log, // (D,N) this layer
            const float* __restrict__ Dp,    // (D)   this layer
            float* __restrict__ y)           // (B,T,D)
{
    const int b = blockIdx.x;
    const int d = blockIdx.y * 64 + threadIdx.x;

    float Arow[NN];
    #pragma unroll
    for (int n = 0; n < NN; ++n) Arow[n] = -expf(A_log[d * NN + n]);
    const float dskip = Dp[d];

    float s[NN];
    #pragma unroll
    for (int n = 0; n < NN; ++n) s[n] = 0.0f;

    __shared__ float sB[NN];
    __shared__ float sC[NN];

    for (int t = 0; t < TT; ++t) {
        const size_t bt = (size_t)b * TT + t;
        if (threadIdx.x < NN)            sB[threadIdx.x]      = Bm[bt * NN + threadIdx.x];
        else if (threadIdx.x < 2 * NN)   sC[threadIdx.x - NN] = Cm[bt * NN + (threadIdx.x - NN)];
        __syncthreads();

        const float xv = xz[bt * (2 * DD) + d];
        const float zv = xz[bt * (2 * DD) + DD + d];
        const float dl = delta[bt * DD + d];
        const float dx = dl * xv;
        float acc = dskip * xv;
        #pragma unroll
        for (int n = 0; n < NN; ++n) {
            const float dA = expf(dl * Arow[n]);
            s[n] = dA * s[n] + dx * sB[n];
            acc += s[n] * sC[n];
        }
        const float sig = 1.0f / (1.0f + expf(-zv));
        y[bt * DD + d] = acc * (zv * sig);
        __syncthreads();
    }
}

// ---------------------------------------------------------------------------
// Attention pooling (masked softmax over T) + Cox head. One block per batch row.
// ---------------------------------------------------------------------------
__global__ void __launch_bounds__(256)
pool_head_kernel(const float* __restrict__ h,    // (B,T,D)
                 const float* __restrict__ mask, // (B,T)
                 const float* __restrict__ Wa, const float* __restrict__ ba,
                 const float* __restrict__ W1, const float* __restrict__ b1,
                 const float* __restrict__ W2, const float* __restrict__ b2,
                 float* __restrict__ risk)       // (B)
{
    __shared__ float attn[TT];
    __shared__ float red[256];
    __shared__ float pooled[DD];
    __shared__ float hidden[DD / 2];

    const int b   = blockIdx.x;
    const int tid = threadIdx.x;
    const float FILL = -3.4028234663852886e38f;   // finfo(f32).min

    // logits
    for (int t = tid; t < TT; t += 256) {
        const float* hp = h + ((size_t)b * TT + t) * DD;
        float dot = ba[0];
        for (int d = 0; d < DD; ++d) dot += hp[d] * Wa[d];
        attn[t] = (mask[(size_t)b * TT + t] == 0.0f) ? FILL : dot;
    }
    __syncthreads();

    // softmax: max
    float lmax = FILL;
    for (int t = tid; t < TT; t += 256) lmax = fmaxf(lmax, attn[t]);
    red[tid] = lmax;
    __syncthreads();
    #pragma unroll
    for (int off = 128; off > 0; off >>= 1) {
        if (tid < off) red[tid] = fmaxf(red[tid], red[tid + off]);
        __syncthreads();
    }
    lmax = red[0];
    __syncthreads();

    // softmax: exp + sum (each thread owns t == tid mod 256)
    float lsum = 0.0f;
    for (int t = tid; t < TT; t += 256) {
        const float e = expf(attn[t] - lmax);
        attn[t] = e;
        lsum += e;
    }
    red[tid] = lsum;
    __syncthreads();
    #pragma unroll
    for (int off = 128; off > 0; off >>= 1) {
        if (tid < off) red[tid] += red[tid + off];
        __syncthreads();
    }
    const float inv = 1.0f / red[0];

    // pooled[d]
    float p = 0.0f;
    for (int t = 0; t < TT; ++t)
        p += h[((size_t)b * TT + t) * DD + tid] * attn[t];
    pooled[tid] = p * inv;
    __syncthreads();

    // hidden = relu(pooled @ W1^T + b1)
    if (tid < DD / 2) {
        float v = b1[tid];
        for (int d = 0; d < DD; ++d) v += pooled[d] * W1[tid * DD + d];
        hidden[tid] = fmaxf(v, 0.0f);
    }
    __syncthreads();

    if (tid == 0) {
        float v = b2[0];
        for (int j = 0; j < DD / 2; ++j) v += hidden[j] * W2[j];
        risk[b] = v;
    }
}

// ---------------------------------------------------------------------------
// Launch
// ---------------------------------------------------------------------------
extern "C" void kernel_launch(void* const* d_in, const int* in_sizes, int n_in,
                              void* d_out, int out_size, void* d_ws, size_t ws_size,
                              hipStream_t stream)
{
    (void)in_sizes; (void)n_in; (void)out_size; (void)ws_size;

    const float* X       = (const float*)d_in[0];
    const float* delta_t = (const float*)d_in[1];
    const float* mask    = (const float*)d_in[2];
    const float* Wp      = (const float*)d_in[3];
    const float* bp      = (const float*)d_in[4];
    const float* g0      = (const float*)d_in[5];
    const float* b0      = (const float*)d_in[6];
    const float* ln_g    = (const float*)d_in[7];
    const float* ln_b    = (const float*)d_in[8];
    const float* Win     = (const float*)d_in[9];
    const float* bin_    = (const float*)d_in[10];
    const float* Wdt     = (const float*)d_in[11];
    const float* bdt     = (const float*)d_in[12];
    const float* WB      = (const float*)d_in[13];
    const float* WC      = (const float*)d_in[14];
    const float* A_log   = (const float*)d_in[15];
    const float* Dp      = (const float*)d_in[16];
    const float* Wout    = (const float*)d_in[17];
    const float* bout    = (const float*)d_in[18];
    const float* Wa      = (const float*)d_in[19];
    const float* ba      = (const float*)d_in[20];
    const float* W1      = (const float*)d_in[21];
    const float* b1      = (const float*)d_in[22];
    const float* W2      = (const float*)d_in[23];
    const float* b2      = (const float*)d_in[24];

    float* ws  = (float*)d_ws;
    size_t off = 0;
    float* h   = ws + off; off += (size_t)BT * DD;       // 16 MB
    float* hn  = ws + off; off += (size_t)BT * DD;       // 16 MB (also proj scratch)
    float* xz  = ws + off; off += (size_t)BT * 2 * DD;   // 32 MB
    float* dl  = ws + off; off += (size_t)BT * DD;       // 16 MB
    float* Bm  = ws + off; off += (size_t)BT * NN;       //  1 MB
    float* Cm  = ws + off; off += (size_t)BT * NN;       //  1 MB
    float* yb  = ws + off; off += (size_t)BT * DD;       // 16 MB

    // 1) input projection (WMMA) -> hn raw, then LN + exact GELU -> h
    gemm_wmma_f32<<<dim3(BT / 16, DD / 16), 32, 0, stream>>>(
        X, DIN, Wp, DIN, bp, nullptr, hn, DD, DIN);
    ln_act_kernel<<<BT, 256, 0, stream>>>(hn, g0, b0, h, /*gelu=*/1);

    // 2) Mamba layers
    for (int l = 0; l < LL; ++l) {
        ln_act_kernel<<<BT, 256, 0, stream>>>(h, ln_g + l * DD, ln_b + l * DD, hn, 0);

        gemm_wmma_f32<<<dim3(BT / 16, (2 * DD) / 16), 32, 0, stream>>>(
            hn, DD, Win + (size_t)l * 2 * DD * DD, DD, bin_ + l * 2 * DD,
            nullptr, xz, 2 * DD, DD);

        delta_kernel<<<(BT * DD) / 256, 256, 0, stream>>>(
            delta_t, Wdt + (size_t)l * DD * DDT, bdt + l * DD, mask, dl);

        gemm_wmma_f32<<<dim3(BT / 16, NN / 16), 32, 0, stream>>>(
            hn, DD, WB + (size_t)l * NN * DD, DD, nullptr, nullptr, Bm, NN, DD);
        gemm_wmma_f32<<<dim3(BT / 16, NN / 16), 32, 0, stream>>>(
            hn, DD, WC + (size_t)l * NN * DD, DD, nullptr, nullptr, Cm, NN, DD);

        scan_kernel<<<dim3(BB, DD / 64), 64, 0, stream>>>(
            xz, dl, Bm, Cm, A_log + (size_t)l * DD * NN, Dp + l * DD, yb);

        // h = h + y @ Wout^T + bout   (residual fused in epilogue, in place)
        gemm_wmma_f32<<<dim3(BT / 16, DD / 16), 32, 0, stream>>>(
            yb, DD, Wout + (size_t)l * DD * DD, DD, bout + l * DD, h, h, DD, DD);
    }

    // 3) attention pooling + Cox head
    pool_head_kernel<<<BB, 256, 0, stream>>>(
        h, mask, Wa, ba, W1, b1, W2, b2, (float*)d_out);
}